// EngramEmbeddings_5179730559118
// MI455X (gfx1250) — compile-verified
//
#include <hip/hip_runtime.h>
#include <hip/hip_bf16.h>
#include <stdint.h>

// EngramEmbeddings: hashed n-gram embedding gather.
// B=8, S=4096, 16 tables (8 bigram + 8 trigram heads), 80 f32 per row,
// out = [B, S, 1280] f32.
//
// Pure bandwidth problem (~168 MB gathered + ~168 MB written, 0 FLOPs).
// Tables total ~185 MB -> resident in the 192 MB L2. We protect that
// residency by writing the output with non-temporal (NT) store hints, and
// move all data as 128-bit vectors.
//
// Two-phase plan:
//   1) index kernel: 1 thread per position computes all 16 int64 hashes and
//      constant-divisor mods (fully occupied lanes, no divergence) -> d_ws.
//   2) gather kernel: pure streaming, no LDS/barriers; idx load (L2 hit),
//      b128 table gather (L2 hit), b128 NT output store.

#define ENG_B 8
#define ENG_S 4096
#define ENG_POS (ENG_B * ENG_S)   // 32768
#define ENG_SLOT 80               // floats per table row
#define ENG_TABLES 16
#define ENG_OUTD 1280             // 16 * 80
#define ENG_BLOCK 320             // 10 wave32: one float4 per thread per position
#define ENG_IDX_BLOCK 256

typedef float v4f __attribute__((ext_vector_type(4)));
typedef uint32_t v4u __attribute__((ext_vector_type(4)));

struct EngramTables {
    const float* w[ENG_TABLES];   // d_in[2 + t], t = 0..15 (n2 h0..7, n3 h0..7)
};

// ---------------------------------------------------------------------------
// Phase 1: per-position hash + constant-divisor mod for all 16 tables.
// ---------------------------------------------------------------------------
__global__ __launch_bounds__(ENG_IDX_BLOCK)
void engram_index_kernel(const long long* __restrict__ ids,     // [B, S] int64
                         const long long* __restrict__ seeds,   // [3, 8] int64
                         uint32_t* __restrict__ idx_out)        // [POS, 16]
{
    const int pos = blockIdx.x * ENG_IDX_BLOCK + threadIdx.x;   // 0 .. 32767
    const int b   = pos >> 12;
    const int s   = pos & (ENG_S - 1);

    const long long* row = ids + (size_t)b * ENG_S;
    const long long id0 = row[s];                               // current token
    const long long id1 = (s >= 1) ? row[s - 1] : 0ll;          // prev (pad 0)
    const long long id2 = (s >= 2) ? row[s - 2] : 0ll;          // prev-prev

    constexpr unsigned long long N2[8] = {
        6619ull, 6637ull, 6653ull, 6659ull, 6661ull, 6673ull, 6679ull, 6689ull };
    constexpr unsigned long long N3[8] = {
        65521ull, 65537ull, 65539ull, 65543ull, 65551ull, 65557ull, 65563ull, 65579ull };

    uint32_t r[ENG_TABLES];

    // bigram heads: h = id[s-1]*seed[0][k] ^ id[s]*seed[1][k]
#pragma unroll
    for (int k = 0; k < 8; ++k) {
        const unsigned long long h =
            (unsigned long long)(id1 * seeds[0 * 8 + k]) ^
            (unsigned long long)(id0 * seeds[1 * 8 + k]);
        r[k] = (uint32_t)(h % N2[k]);       // literal divisor -> mul-hi magic
    }
    // trigram heads: h = id[s-2]*s0 ^ id[s-1]*s1 ^ id[s]*s2
#pragma unroll
    for (int k = 0; k < 8; ++k) {
        const unsigned long long h =
            (unsigned long long)(id2 * seeds[0 * 8 + k]) ^
            (unsigned long long)(id1 * seeds[1 * 8 + k]) ^
            (unsigned long long)(id0 * seeds[2 * 8 + k]);
        r[8 + k] = (uint32_t)(h % N3[k]);
    }

    // 64 B per position, written as 4x b128 (stays hot in L2 for phase 2).
    v4u* dst = (v4u*)(idx_out + (size_t)pos * ENG_TABLES);
#pragma unroll
    for (int q = 0; q < 4; ++q) {
        v4u v = { r[4 * q + 0], r[4 * q + 1], r[4 * q + 2], r[4 * q + 3] };
        dst[q] = v;
    }
}

// ---------------------------------------------------------------------------
// Phase 2: pure streaming gather. 1 block = 1 position = 320 float4 lanes.
// ---------------------------------------------------------------------------
__global__ __launch_bounds__(ENG_BLOCK)
void engram_gather_kernel(const uint32_t* __restrict__ idx_in,  // [POS, 16]
                          EngramTables tabs,
                          float* __restrict__ out)              // [B, S, 1280]
{
    const int pos    = blockIdx.x;
    const int t      = threadIdx.x;
    const int tab    = t / 20;            // table id 0..15
    const int within = (t % 20) * 4;      // float offset inside the 80-f row

    const uint32_t row = idx_in[(size_t)pos * ENG_TABLES + tab];   // L2 hit

    // 320 B rows are 16 B aligned -> global_load_b128 from L2-resident table.
    const v4f val = *(const v4f*)(tabs.w[tab] + (size_t)row * ENG_SLOT + within);

    // NT store (th:TH_STORE_NT): stream the 168 MB output past L2 so the
    // ~185 MB of tables stay resident in the 192 MB L2.
    v4f* dst = (v4f*)(out + (size_t)pos * ENG_OUTD) + t;
    __builtin_nontemporal_store(val, dst);
}

// ---------------------------------------------------------------------------
// Fallback: fused single-kernel path (used only if d_ws is too small).
// ---------------------------------------------------------------------------
__global__ __launch_bounds__(ENG_BLOCK)
void engram_fused_kernel(const long long* __restrict__ ids,
                         const long long* __restrict__ seeds,
                         EngramTables tabs,
                         float* __restrict__ out)
{
    constexpr unsigned long long SIZES[ENG_TABLES] = {
        6619ull, 6637ull, 6653ull, 6659ull, 6661ull, 6673ull, 6679ull, 6689ull,
        65521ull, 65537ull, 65539ull, 65543ull, 65551ull, 65557ull, 65563ull, 65579ull
    };

    __shared__ uint32_t s_idx[ENG_TABLES];

    const int pos = blockIdx.x;
    const int b   = pos >> 12;
    const int s   = pos & (ENG_S - 1);
    const int t   = threadIdx.x;

    if (t < ENG_TABLES) {
        const long long* row = ids + (size_t)b * ENG_S;
        const long long id0 = row[s];
        const long long id1 = (s >= 1) ? row[s - 1] : 0ll;
        const long long id2 = (s >= 2) ? row[s - 2] : 0ll;

        const int head = t & 7;
        unsigned long long h;
        if (t < 8) {
            h = (unsigned long long)(id1 * seeds[0 * 8 + head])
              ^ (unsigned long long)(id0 * seeds[1 * 8 + head]);
        } else {
            h = (unsigned long long)(id2 * seeds[0 * 8 + head])
              ^ (unsigned long long)(id1 * seeds[1 * 8 + head])
              ^ (unsigned long long)(id0 * seeds[2 * 8 + head]);
        }

        uint32_t idx = 0;
#pragma unroll
        for (int i = 0; i < ENG_TABLES; ++i) {
            if (t == i) idx = (uint32_t)(h % SIZES[i]);
        }
        s_idx[t] = idx;
    }
    __syncthreads();

    const int tab    = t / 20;
    const int within = (t % 20) * 4;
    const uint32_t idx = s_idx[tab];

    const v4f val = *(const v4f*)(tabs.w[tab] + (size_t)idx * ENG_SLOT + within);
    v4f* dst = (v4f*)(out + (size_t)pos * ENG_OUTD) + t;
    __builtin_nontemporal_store(val, dst);
}

extern "C" void kernel_launch(void* const* d_in, const int* in_sizes, int n_in,
                              void* d_out, int out_size, void* d_ws, size_t ws_size,
                              hipStream_t stream) {
    (void)in_sizes; (void)n_in; (void)out_size;

    const long long* ids   = (const long long*)d_in[0];   // int64 [8,4096]
    const long long* seeds = (const long long*)d_in[1];   // int64 [3,8]

    EngramTables tabs;
#pragma unroll
    for (int i = 0; i < ENG_TABLES; ++i) {
        tabs.w[i] = (const float*)d_in[2 + i];            // n2 h0..7, n3 h0..7
    }

    float* out = (float*)d_out;
    const size_t idx_bytes = (size_t)ENG_POS * ENG_TABLES * sizeof(uint32_t); // 2 MB

    if (d_ws != nullptr && ws_size >= idx_bytes) {
        uint32_t* idx_ws = (uint32_t*)d_ws;
        engram_index_kernel<<<dim3(ENG_POS / ENG_IDX_BLOCK), dim3(ENG_IDX_BLOCK),
                              0, stream>>>(ids, seeds, idx_ws);
        engram_gather_kernel<<<dim3(ENG_POS), dim3(ENG_BLOCK), 0, stream>>>(
            idx_ws, tabs, out);
    } else {
        engram_fused_kernel<<<dim3(ENG_POS), dim3(ENG_BLOCK), 0, stream>>>(
            ids, seeds, tabs, out);
    }
}